// PolicyHead_13855564497497
// MI455X (gfx1250) — compile-verified
//
#include <hip/hip_runtime.h>
#include <stdint.h>

typedef float v2f __attribute__((ext_vector_type(2)));
typedef float v8f __attribute__((ext_vector_type(8)));

#define H      4096
#define G3     12288      // 3*H
#define VOCAB  4
#define NSTEPS 48
#define PADTOK 3

// ---- workspace layout (bytes) ----
#define OFF_EW    0u                      // 4  x 12288 f32  (emb @ W_ih^T)
#define OFF_PW    196608u                 // 48 x 12288 f32  (pos @ W_ih^T)
#define OFF_H     2555904u                // 4096 f32 hidden state
#define OFF_GH    2572288u                // 12288 f32 per-step W_hh gates
#define OFF_TOK   2621440u                // int current input token
#define OFF_WHHB  2621696u                // 12288*4096 bf16 (optional)
#define WS_NEED_BF16 (OFF_WHHB + (size_t)G3 * H * 2u)

// ---------------------------------------------------------------------------
// init: h = tensor_state, tok = PAD, write the 4 interleaved PAD outputs
// ---------------------------------------------------------------------------
__global__ void gru_init_kernel(const float* __restrict__ tensor_state,
                                float* __restrict__ h, int* __restrict__ tok,
                                int* __restrict__ out) {
  int tid = threadIdx.x;
  for (int i = tid; i < H; i += 256) h[i] = tensor_state[i];
  if (tid == 0) {
    *tok = PADTOK;
    out[12] = PADTOK; out[25] = PADTOK; out[38] = PADTOK; out[51] = PADTOK;
  }
}

// ---------------------------------------------------------------------------
// one-time fp32 -> bf16 (RNE) conversion of W_hh so the serial loop is
// L2-resident (100.7 MB < 192 MB global L2). fp32 source is streamed once ->
// non-temporal so it does not evict the bf16 copy we are building.
// ---------------------------------------------------------------------------
__global__ void whh_to_bf16_kernel(const float* __restrict__ src,
                                   unsigned short* __restrict__ dst, int n) {
  for (int i = blockIdx.x * blockDim.x + threadIdx.x; i < n;
       i += gridDim.x * blockDim.x) {
    float x = __builtin_nontemporal_load(src + i);
    uint32_t u = __float_as_uint(x);
    uint32_t r = u + 0x7FFFu + ((u >> 16) & 1u);   // round-to-nearest-even
    dst[i] = (unsigned short)(r >> 16);
  }
}

// ---------------------------------------------------------------------------
// Phase-1 GEMM with fp32 WMMA: D(64 x 12288) = A(64 x 4096) * W_ih^T
//   A rows 0..3   = token_embedding (rows 4..15 replicated, never stored)
//   A rows 16..63 = pos_embedding[0..47]
// One wave per 16-wide N tile; it carries all 4 M tiles so W_ih streams once
// (non-temporal: W_ih is dead after this kernel; keep L2 for bf16 W_hh).
// ---------------------------------------------------------------------------
__global__ void gi_gemm_wmma_kernel(const float* __restrict__ tokEmb,
                                    const float* __restrict__ posEmb,
                                    const float* __restrict__ W_ih,
                                    float* __restrict__ EW,
                                    float* __restrict__ PW) {
  const int wave = (blockIdx.x * blockDim.x + threadIdx.x) >> 5;
  const int nTiles = G3 / 16;                 // 768
  if (wave >= nTiles) return;                 // wave-uniform guard
  const int n0   = wave * 16;
  const int lane = threadIdx.x & 31;
  const int half = lane >> 4;                 // 0: K=0,1   1: K=2,3
  const int mrow = lane & 15;

  // A row base pointers per M tile (layout: 16x4 fp32 A, v2f per lane).
  // Token tile: replicate rows via (mrow & 3) -> no EXEC divergence in the
  // loop; garbage rows M=4..15 are finite and never stored.
  const float* a0p = tokEmb + (size_t)(mrow & 3) * H;
  const float* a1p = posEmb + (size_t)(0 * 16 + mrow) * H;   // pos rows 0..15
  const float* a2p = posEmb + (size_t)(1 * 16 + mrow) * H;   // pos rows 16..31
  const float* a3p = posEmb + (size_t)(2 * 16 + mrow) * H;   // pos rows 32..47
  const float* bp  = W_ih  + (size_t)(n0 + mrow) * H;        // B[k][n]=W_ih[n][k]

  v8f acc[4] = {};
  for (int k0 = 0; k0 < H; k0 += 4) {
    const int k = k0 + half * 2;
    v2f b  = __builtin_nontemporal_load((const v2f*)(bp + k));
    v2f a0 = *(const v2f*)(a0p + k);
    v2f a1 = *(const v2f*)(a1p + k);
    v2f a2 = *(const v2f*)(a2p + k);
    v2f a3 = *(const v2f*)(a3p + k);
    acc[0] = __builtin_amdgcn_wmma_f32_16x16x4_f32(false, a0, false, b,
                                                   (short)0, acc[0], false, false);
    acc[1] = __builtin_amdgcn_wmma_f32_16x16x4_f32(false, a1, false, b,
                                                   (short)0, acc[1], false, false);
    acc[2] = __builtin_amdgcn_wmma_f32_16x16x4_f32(false, a2, false, b,
                                                   (short)0, acc[2], false, false);
    acc[3] = __builtin_amdgcn_wmma_f32_16x16x4_f32(false, a3, false, b,
                                                   (short)0, acc[3], false, false);
  }

  // C/D layout: VGPR r -> M = r + 8*half (lanes 0-15 / 16-31), N = n0 + mrow
  const int ncol = n0 + mrow;
#pragma unroll
  for (int mt = 0; mt < 4; ++mt) {
#pragma unroll
    for (int r = 0; r < 8; ++r) {
      const int M = mt * 16 + r + half * 8;
      const float val = acc[mt][r];
      if (M < VOCAB)                 EW[(size_t)M * G3 + ncol] = val;
      else if (M >= 16 && M < 64)    PW[(size_t)(M - 16) * G3 + ncol] = val;
    }
  }
}

// ---------------------------------------------------------------------------
// serial matvec: gh[n] = sum_k W_hh[n][k] * h[k] + b_hh[n]   (n = 0..12287)
// one wave per output row; h staged in LDS; bf16 path keeps W_hh in L2
// ---------------------------------------------------------------------------
__global__ void gru_matvec_kernel(const unsigned short* __restrict__ WhhB,
                                  const float* __restrict__ WhhF,
                                  const float* __restrict__ b_hh,
                                  const float* __restrict__ h,
                                  float* __restrict__ gh, int use_bf16) {
  __shared__ float sh[H];
  for (int i = threadIdx.x; i < H; i += 256) sh[i] = h[i];
  __syncthreads();

  const int n    = blockIdx.x * 8 + (threadIdx.x >> 5);
  const int lane = threadIdx.x & 31;
  float acc = 0.f;

  if (use_bf16) {
    const ushort4* row = (const ushort4*)(WhhB + (size_t)n * H);
#pragma unroll 4
    for (int i = 0; i < H / 128; ++i) {                 // 32 iters
      const int base = i * 128 + lane * 4;
      ushort4 w = row[i * 32 + lane];                   // 8B coalesced
      __builtin_prefetch(&row[i * 32 + lane + 256], 0, 1);
      acc = fmaf(__uint_as_float((uint32_t)w.x << 16), sh[base + 0], acc);
      acc = fmaf(__uint_as_float((uint32_t)w.y << 16), sh[base + 1], acc);
      acc = fmaf(__uint_as_float((uint32_t)w.z << 16), sh[base + 2], acc);
      acc = fmaf(__uint_as_float((uint32_t)w.w << 16), sh[base + 3], acc);
    }
  } else {
    const float4* row = (const float4*)(WhhF + (size_t)n * H);
#pragma unroll 4
    for (int i = 0; i < H / 128; ++i) {
      const int base = i * 128 + lane * 4;
      float4 w = row[i * 32 + lane];                    // 16B coalesced
      __builtin_prefetch(&row[i * 32 + lane + 128], 0, 1);
      acc = fmaf(w.x, sh[base + 0], acc);
      acc = fmaf(w.y, sh[base + 1], acc);
      acc = fmaf(w.z, sh[base + 2], acc);
      acc = fmaf(w.w, sh[base + 3], acc);
    }
  }
#pragma unroll
  for (int off = 16; off > 0; off >>= 1) acc += __shfl_xor(acc, off, 32);
  if (lane == 0) gh[n] = acc + b_hh[n];
}

// ---------------------------------------------------------------------------
// gate math: h = (1-z)*tanh(gi_n + r*gh_n) + z*h   (gi = EW[tok]+PW[t]+b_ih)
// ---------------------------------------------------------------------------
__global__ void gru_update_kernel(const float* __restrict__ EW,
                                  const float* __restrict__ PW,
                                  const float* __restrict__ b_ih,
                                  const float* __restrict__ gh,
                                  float* __restrict__ h,
                                  const int* __restrict__ tok, int t) {
  const int n = blockIdx.x * blockDim.x + threadIdx.x;
  if (n >= H) return;
  const int tk = *tok;
  const float* ew = EW + (size_t)tk * G3;
  const float* pw = PW + (size_t)t * G3;
  const float gi_r = ew[n]            + pw[n]            + b_ih[n];
  const float gi_z = ew[n + H]        + pw[n + H]        + b_ih[n + H];
  const float gi_n = ew[n + 2 * H]    + pw[n + 2 * H]    + b_ih[n + 2 * H];
  const float gh_r = gh[n];
  const float gh_z = gh[n + H];
  const float gh_n = gh[n + 2 * H];
  const float r = 1.f / (1.f + expf(-(gi_r + gh_r)));
  const float z = 1.f / (1.f + expf(-(gi_z + gh_z)));
  const float nn = tanhf(gi_n + r * gh_n);
  h[n] = (1.f - z) * nn + z * h[n];
}

// ---------------------------------------------------------------------------
// logits = W_out @ h + b_out; argmax (first max); emit token, set next input
// ---------------------------------------------------------------------------
__global__ void gru_logits_kernel(const float* __restrict__ W_out,
                                  const float* __restrict__ b_out,
                                  const float* __restrict__ h,
                                  int* __restrict__ tok,
                                  int* __restrict__ out, int t) {
  __shared__ float red[VOCAB][256];
  const int tid = threadIdx.x;
  float acc[VOCAB] = {0.f, 0.f, 0.f, 0.f};
  for (int k = tid; k < H; k += 256) {
    const float hv = h[k];
#pragma unroll
    for (int v = 0; v < VOCAB; ++v) acc[v] = fmaf(W_out[v * H + k], hv, acc[v]);
  }
#pragma unroll
  for (int v = 0; v < VOCAB; ++v) red[v][tid] = acc[v];
  __syncthreads();
  for (int s = 128; s > 0; s >>= 1) {
    if (tid < s) {
#pragma unroll
      for (int v = 0; v < VOCAB; ++v) red[v][tid] += red[v][tid + s];
    }
    __syncthreads();
  }
  if (tid == 0) {
    int best = 0;
    float bv = red[0][0] + b_out[0];
    for (int v = 1; v < VOCAB; ++v) {
      const float lv = red[v][0] + b_out[v];
      if (lv > bv) { bv = lv; best = v; }      // first-max semantics
    }
    out[(t / 12) * 13 + (t % 12)] = best;
    *tok = (((t + 1) % 12) == 0) ? PADTOK : best;
  }
}

// ---------------------------------------------------------------------------
extern "C" void kernel_launch(void* const* d_in, const int* in_sizes, int n_in,
                              void* d_out, int out_size, void* d_ws,
                              size_t ws_size, hipStream_t stream) {
  (void)in_sizes; (void)n_in; (void)out_size;
  const float* tensor_state = (const float*)d_in[0];
  const float* tokEmb       = (const float*)d_in[1];
  const float* posEmb       = (const float*)d_in[2];
  const float* W_ih         = (const float*)d_in[3];
  const float* W_hh         = (const float*)d_in[4];
  const float* b_ih         = (const float*)d_in[5];
  const float* b_hh         = (const float*)d_in[6];
  const float* W_out        = (const float*)d_in[7];
  const float* b_out        = (const float*)d_in[8];
  int* out = (int*)d_out;

  char* ws = (char*)d_ws;
  float* EW  = (float*)(ws + OFF_EW);
  float* PW  = (float*)(ws + OFF_PW);
  float* h   = (float*)(ws + OFF_H);
  float* gh  = (float*)(ws + OFF_GH);
  int*   tok = (int*)  (ws + OFF_TOK);
  unsigned short* WhhB = (unsigned short*)(ws + OFF_WHHB);
  const int use_bf16 = (ws_size >= WS_NEED_BF16) ? 1 : 0;

  gru_init_kernel<<<1, 256, 0, stream>>>(tensor_state, h, tok, out);
  // GEMM first (streams W_ih with NT hints), then the bf16 conversion, so
  // W_hh_bf16 is the freshest content in L2 when the serial loop starts.
  gi_gemm_wmma_kernel<<<96, 256, 0, stream>>>(tokEmb, posEmb, W_ih, EW, PW);
  if (use_bf16)
    whh_to_bf16_kernel<<<2048, 256, 0, stream>>>(W_hh, WhhB, G3 * H);

  for (int t = 0; t < NSTEPS; ++t) {
    gru_matvec_kernel<<<G3 / 8, 256, 0, stream>>>(WhhB, W_hh, b_hh, h, gh,
                                                  use_bf16);
    gru_update_kernel<<<H / 256, 256, 0, stream>>>(EW, PW, b_ih, gh, h, tok, t);
    gru_logits_kernel<<<1, 256, 0, stream>>>(W_out, b_out, h, tok, out, t);
  }
}